// GPTModel_29944511987986
// MI455X (gfx1250) — compile-verified
//
#include <hip/hip_runtime.h>
#include <math.h>

typedef __attribute__((ext_vector_type(2))) float v2f;
typedef __attribute__((ext_vector_type(8))) float v8f;

constexpr int V = 32000, D = 768, H = 12, L = 6, T = 2048, F = 3072;
constexpr int DH = D / H;            // 64
constexpr float EPS = 1e-5f;

// ---------------------------------------------------------------- reductions
__device__ __forceinline__ float block_reduce_sum(float v, float* red) {
  int tid = threadIdx.x;
  red[tid] = v; __syncthreads();
  for (int off = 128; off > 0; off >>= 1) {
    if (tid < off) red[tid] += red[tid + off];
    __syncthreads();
  }
  float r = red[0]; __syncthreads();
  return r;
}
__device__ __forceinline__ float block_reduce_max(float v, float* red) {
  int tid = threadIdx.x;
  red[tid] = v; __syncthreads();
  for (int off = 128; off > 0; off >>= 1) {
    if (tid < off) red[tid] = fmaxf(red[tid], red[tid + off]);
    __syncthreads();
  }
  float r = red[0]; __syncthreads();
  return r;
}

// ---------------------------------------------------------------- embedding
__global__ void embed_kernel(const int* __restrict__ x,
                             const float* __restrict__ tok,
                             const float* __restrict__ pos,
                             float* __restrict__ h) {
  int idx = blockIdx.x * blockDim.x + threadIdx.x;
  if (idx >= T * D) return;
  int t = idx / D, d = idx % D;
  h[idx] = tok[(size_t)x[t] * D + d] + pos[idx];
}

// ---------------------------------------------------------------- layernorm
template <int NCOL>
__global__ void layernorm_kernel(const float* __restrict__ in,
                                 const float* __restrict__ g,
                                 const float* __restrict__ b,
                                 float* __restrict__ out) {
  __shared__ float red[256];
  int row = blockIdx.x;
  const float* r = in + (size_t)row * NCOL;
  float s = 0.f;
  for (int j = threadIdx.x; j < NCOL; j += 256) s += r[j];
  float mean = block_reduce_sum(s, red) / NCOL;
  float sv = 0.f;
  for (int j = threadIdx.x; j < NCOL; j += 256) { float d0 = r[j] - mean; sv += d0 * d0; }
  float rstd = rsqrtf(block_reduce_sum(sv, red) / NCOL + EPS);
  for (int j = threadIdx.x; j < NCOL; j += 256)
    out[(size_t)row * NCOL + j] = (r[j] - mean) * rstd * g[j] + b[j];
}

// ---------------------------------------------------------------- WMMA GEMM (NT)
// C[M,N] = A[M,K] * W[N,K]^T (+bias)(GELU)(+res).
// One wave computes a 32x64 tile: 2 (M) x 4 (N) accumulators of 16x16.
// f32 WMMA 16x16x4: A/B lane<16 -> K={k,k+1}; lane>=16 -> K={k+2,k+3}.
template <bool BIAS, bool GELU, bool RES>
__global__ void gemm_nt_wmma(const float* __restrict__ A, int lda,
                             const float* __restrict__ W, int ldw,
                             const float* __restrict__ bias,
                             const float* __restrict__ res, int ldr,
                             float* __restrict__ C, int ldc, int K) {
  const int lane = threadIdx.x;
  const int half = lane >> 4;
  const int l16  = lane & 15;
  const size_t m0 = (size_t)blockIdx.y * 32;
  const size_t n0 = (size_t)blockIdx.x * 64;

  const float* ap0 = A + (m0 + l16) * (size_t)lda + half * 2;
  const float* ap1 = ap0 + 16 * (size_t)lda;
  const float* wp0 = W + (n0 + l16) * (size_t)ldw + half * 2;
  const float* wp1 = wp0 + 16 * (size_t)ldw;
  const float* wp2 = wp1 + 16 * (size_t)ldw;
  const float* wp3 = wp2 + 16 * (size_t)ldw;

  v8f acc[2][4] = {};
  for (int k0 = 0; k0 < K; k0 += 16) {
    __builtin_prefetch(ap0 + 128, 0, 3);   // global_prefetch_b8
    __builtin_prefetch(wp0 + 128, 0, 3);
#pragma unroll
    for (int kk = 0; kk < 16; kk += 4) {
      v2f a0 = *(const v2f*)(ap0 + kk);
      v2f a1 = *(const v2f*)(ap1 + kk);
      v2f b0 = *(const v2f*)(wp0 + kk);
      v2f b1 = *(const v2f*)(wp1 + kk);
      v2f b2 = *(const v2f*)(wp2 + kk);
      v2f b3 = *(const v2f*)(wp3 + kk);
      acc[0][0] = __builtin_amdgcn_wmma_f32_16x16x4_f32(false, a0, false, b0, (short)0, acc[0][0], false, false);
      acc[0][1] = __builtin_amdgcn_wmma_f32_16x16x4_f32(false, a0, false, b1, (short)0, acc[0][1], false, false);
      acc[0][2] = __builtin_amdgcn_wmma_f32_16x16x4_f32(false, a0, false, b2, (short)0, acc[0][2], false, false);
      acc[0][3] = __builtin_amdgcn_wmma_f32_16x16x4_f32(false, a0, false, b3, (short)0, acc[0][3], false, false);
      acc[1][0] = __builtin_amdgcn_wmma_f32_16x16x4_f32(false, a1, false, b0, (short)0, acc[1][0], false, false);
      acc[1][1] = __builtin_amdgcn_wmma_f32_16x16x4_f32(false, a1, false, b1, (short)0, acc[1][1], false, false);
      acc[1][2] = __builtin_amdgcn_wmma_f32_16x16x4_f32(false, a1, false, b2, (short)0, acc[1][2], false, false);
      acc[1][3] = __builtin_amdgcn_wmma_f32_16x16x4_f32(false, a1, false, b3, (short)0, acc[1][3], false, false);
    }
    ap0 += 16; ap1 += 16; wp0 += 16; wp1 += 16; wp2 += 16; wp3 += 16;
  }

#pragma unroll
  for (int nj = 0; nj < 4; ++nj) {
    size_t col = n0 + nj * 16 + l16;
    float bv = BIAS ? bias[col] : 0.f;
#pragma unroll
    for (int mi = 0; mi < 2; ++mi) {
#pragma unroll
      for (int i = 0; i < 8; ++i) {
        size_t row = m0 + mi * 16 + i + half * 8;  // C VGPR i: lanes0-15 M=i, lanes16-31 M=i+8
        float v = acc[mi][nj][i] + bv;
        if (GELU) v = 0.5f * v * (1.f + erff(v * 0.70710678118654752f));
        if (RES)  v += res[row * (size_t)ldr + col];
        C[row * (size_t)ldc + col] = v;
      }
    }
  }
}

// ---------------------------------------------------------------- WMMA GEMM (NN)
// C[M,N] = A[M,K] * B[K,N]  (B row-major, for ctx = P @ V). 32x64 tile per wave.
__global__ void gemm_nn_wmma(const float* __restrict__ A, int lda,
                             const float* __restrict__ Bm, int ldb,
                             float* __restrict__ C, int ldc, int K) {
  const int lane = threadIdx.x;
  const int half = lane >> 4;
  const int l16  = lane & 15;
  const size_t m0 = (size_t)blockIdx.y * 32;
  const size_t n0 = (size_t)blockIdx.x * 64;

  const float* ap0 = A + (m0 + l16) * (size_t)lda + half * 2;
  const float* ap1 = ap0 + 16 * (size_t)lda;
  const float* bp0 = Bm + (size_t)(half * 2) * ldb + n0 + l16;  // row k+half*2
  const float* bp1 = bp0 + (size_t)ldb;                          // row k+half*2+1

  v8f acc[2][4] = {};
  for (int k0 = 0; k0 < K; k0 += 16) {
    __builtin_prefetch(ap0 + 128, 0, 3);
#pragma unroll
    for (int kk = 0; kk < 16; kk += 4) {
      v2f a0 = *(const v2f*)(ap0 + kk);
      v2f a1 = *(const v2f*)(ap1 + kk);
      size_t roff = (size_t)kk * ldb;
#pragma unroll
      for (int nj = 0; nj < 4; ++nj) {
        v2f b;
        b.x = bp0[roff + nj * 16];
        b.y = bp1[roff + nj * 16];
        acc[0][nj] = __builtin_amdgcn_wmma_f32_16x16x4_f32(false, a0, false, b, (short)0, acc[0][nj], false, false);
        acc[1][nj] = __builtin_amdgcn_wmma_f32_16x16x4_f32(false, a1, false, b, (short)0, acc[1][nj], false, false);
      }
    }
    ap0 += 16; ap1 += 16;
    bp0 += (size_t)16 * ldb; bp1 += (size_t)16 * ldb;
  }

#pragma unroll
  for (int nj = 0; nj < 4; ++nj) {
    size_t col = n0 + nj * 16 + l16;
#pragma unroll
    for (int mi = 0; mi < 2; ++mi) {
#pragma unroll
      for (int i = 0; i < 8; ++i) {
        size_t row = m0 + mi * 16 + i + half * 8;
        C[row * (size_t)ldc + col] = acc[mi][nj][i];
      }
    }
  }
}

// ---------------------------------------------------------------- softmax (causal, scaled)
__global__ void softmax_causal_kernel(float* __restrict__ s) {
  __shared__ float red[256];
  int i = blockIdx.x;                    // query row
  float* row = s + (size_t)i * T;
  const float scale = 0.125f;            // 1/sqrt(64)
  float m = -3.4e38f;
  for (int j = threadIdx.x; j <= i; j += 256) m = fmaxf(m, row[j] * scale);
  m = block_reduce_max(m, red);
  float sum = 0.f;
  for (int j = threadIdx.x; j < T; j += 256) {
    if (j <= i) { float e = __expf(row[j] * scale - m); row[j] = e; sum += e; }
    else row[j] = 0.f;
  }
  float inv = 1.f / block_reduce_sum(sum, red);
  for (int j = threadIdx.x; j <= i; j += 256) row[j] *= inv;
}

// ---------------------------------------------------------------- cross entropy
__global__ void zero2_kernel(float* a) { if (threadIdx.x < 2) a[threadIdx.x] = 0.f; }

__global__ void ce_kernel(const float* __restrict__ logits,
                          const int* __restrict__ tgt,
                          float* __restrict__ acc2) {
  __shared__ float red[256];
  int t = blockIdx.x;
  const float* row = logits + (size_t)t * V;
  float m = -3.4e38f;
  for (int j = threadIdx.x; j < V; j += 256) m = fmaxf(m, row[j]);
  m = block_reduce_max(m, red);
  float s = 0.f;
  for (int j = threadIdx.x; j < V; j += 256) s += __expf(row[j] - m);
  s = block_reduce_sum(s, red);
  if (threadIdx.x == 0) {
    int y = tgt[t];
    if (y != 0) {
      float nll = -(row[y] - m - __logf(s));
      atomicAdd(&acc2[0], nll);
      atomicAdd(&acc2[1], 1.0f);
    }
  }
}

__global__ void finalize_loss_kernel(const float* __restrict__ acc2,
                                     float* __restrict__ loss) {
  *loss = acc2[0] / fmaxf(acc2[1], 1.0f);
}

// ---------------------------------------------------------------- launch
extern "C" void kernel_launch(void* const* d_in, const int* in_sizes, int n_in,
                              void* d_out, int out_size, void* d_ws, size_t ws_size,
                              hipStream_t stream) {
  const int*   x       = (const int*)  d_in[0];
  const int*   targets = (const int*)  d_in[1];
  const float* tok_emb = (const float*)d_in[2];
  const float* pos_emb = (const float*)d_in[3];
  const float* Wqkv    = (const float*)d_in[4];
  const float* Wo      = (const float*)d_in[5];
  const float* ln1_g   = (const float*)d_in[6];
  const float* ln1_b   = (const float*)d_in[7];
  const float* ln2_g   = (const float*)d_in[8];
  const float* ln2_b   = (const float*)d_in[9];
  const float* W1      = (const float*)d_in[10];
  const float* b1      = (const float*)d_in[11];
  const float* W2      = (const float*)d_in[12];
  const float* b2      = (const float*)d_in[13];
  const float* lnf_g   = (const float*)d_in[14];
  const float* lnf_b   = (const float*)d_in[15];
  float* out = (float*)d_out;

  // workspace partition (floats)
  float* h    = (float*)d_ws;
  float* hn   = h   + (size_t)T * D;
  float* qkv  = hn  + (size_t)T * D;
  float* ctx  = qkv + (size_t)T * 3 * D;
  float* sc   = ctx + (size_t)T * D;
  float* ff   = sc  + (size_t)T * T;
  float* acc2 = ff  + (size_t)T * F;

  dim3 wave(32);

  embed_kernel<<<(T * D + 255) / 256, 256, 0, stream>>>(x, tok_emb, pos_emb, h);

  for (int l = 0; l < L; ++l) {
    const float* wqkv_l = Wqkv + (size_t)l * 3 * D * D;
    const float* wo_l   = Wo   + (size_t)l * D * D;
    const float* w1_l   = W1   + (size_t)l * F * D;
    const float* b1_l   = b1   + (size_t)l * F;
    const float* w2_l   = W2   + (size_t)l * D * F;
    const float* b2_l   = b2   + (size_t)l * D;

    // --- attention sublayer ---
    layernorm_kernel<D><<<T, 256, 0, stream>>>(h, ln1_g + (size_t)l * D,
                                               ln1_b + (size_t)l * D, hn);
    gemm_nt_wmma<false, false, false><<<dim3(3 * D / 64, T / 32), wave, 0, stream>>>(
        hn, D, wqkv_l, D, nullptr, nullptr, 0, qkv, 3 * D, D);

    for (int hh = 0; hh < H; ++hh) {
      const float* qh = qkv + hh * DH;             // [T, DH] stride 3D
      const float* kh = qkv + D + hh * DH;
      const float* vh = qkv + 2 * D + hh * DH;
      // scores = Q @ K^T
      gemm_nt_wmma<false, false, false><<<dim3(T / 64, T / 32), wave, 0, stream>>>(
          qh, 3 * D, kh, 3 * D, nullptr, nullptr, 0, sc, T, DH);
      softmax_causal_kernel<<<T, 256, 0, stream>>>(sc);
      // ctx_head = P @ V
      gemm_nn_wmma<<<dim3(DH / 64, T / 32), wave, 0, stream>>>(
          sc, T, vh, 3 * D, ctx + hh * DH, D, T);
    }
    // h = h + ctx @ Wo^T
    gemm_nt_wmma<false, false, true><<<dim3(D / 64, T / 32), wave, 0, stream>>>(
        ctx, D, wo_l, D, nullptr, h, D, h, D, D);

    // --- FFN sublayer ---
    layernorm_kernel<D><<<T, 256, 0, stream>>>(h, ln2_g + (size_t)l * D,
                                               ln2_b + (size_t)l * D, hn);
    gemm_nt_wmma<true, true, false><<<dim3(F / 64, T / 32), wave, 0, stream>>>(
        hn, D, w1_l, D, b1_l, nullptr, 0, ff, F, D);
    gemm_nt_wmma<true, false, true><<<dim3(D / 64, T / 32), wave, 0, stream>>>(
        ff, F, w2_l, F, b2_l, h, D, h, D, F);
  }

  // final LN + tied LM head -> logits in d_out
  layernorm_kernel<D><<<T, 256, 0, stream>>>(h, lnf_g, lnf_b, hn);
  gemm_nt_wmma<false, false, false><<<dim3(V / 64, T / 32), wave, 0, stream>>>(
      hn, D, tok_emb, D, nullptr, nullptr, 0, out, V, D);

  // cross entropy (ignore_index = 0)
  zero2_kernel<<<1, 32, 0, stream>>>(acc2);
  ce_kernel<<<T, 256, 0, stream>>>(out, targets, acc2);
  finalize_loss_kernel<<<1, 1, 0, stream>>>(acc2, out + (size_t)T * V);
}